// RPNStructure_40484361732758
// MI455X (gfx1250) — compile-verified
//
#include <hip/hip_runtime.h>
#include <hip/hip_bf16.h>
#include <stdint.h>

// ---------------------------------------------------------------------------
// RPN proposal generation for MI455X (gfx1250, wave32).
// Pipeline: radix top-k per (image,level) -> decode/clip -> bitonic sort ->
// greedy NMS (boxes staged to LDS via async global->LDS) -> compact to 1000.
// CDNA5 paths: v_wmma_f32_16x16x4_f32 (tensor-pipe histogram reduction),
// global_load_async_to_lds_b128 / s_wait_asynccnt (NMS LDS staging),
// global_prefetch_b8 (streaming histogram passes), wave32 ballot/ds atomics.
// ---------------------------------------------------------------------------

#define IMG_WH   1024.0f
#define NEGV     (-1e30f)
#define BBOX_CLIP 4.135166556742356f   // log(1000/16)
#define NMS_TH   0.7f
#define MIN_SZ   0.001f
#define A_TOTAL  261120
#define N_B      8
#define K_LVL    1000
#define K_IMG    4000
#define N_PAD    4096
#define LVL_OFF  1025.0f               // max(IMG_H,IMG_W)+1

// level boundaries (3 aspects * (1024/s)^2)
#define L0_END 196608
#define L1_END 245760
#define L2_END 258048

typedef float v2f __attribute__((ext_vector_type(2)));
typedef float v8f __attribute__((ext_vector_type(8)));

#if defined(__gfx1250__) && __has_builtin(__builtin_amdgcn_wmma_f32_16x16x4_f32)
#define HAVE_WMMA_F32X4 1
#endif

struct SelState {
  unsigned prefix;   // accumulated high digits of threshold key
  unsigned above;    // count of keys strictly above threshold so far
  unsigned T;        // final 32-bit threshold key
  unsigned needEq;   // how many key==T to take
  unsigned aboveCtr; // compaction counter (strict above)
  unsigned eqCtr;    // compaction counter (equal)
  unsigned pad0, pad1;
};

// workspace layout (bytes)
#define WS_HIST   0u                          // u32[3][32][4096]
#define WS_STATE  1572864u                    // SelState[32]
#define WS_CSCORE 1573888u                    // f32[32][1000]
#define WS_CIDX   1701888u                    // u32[32][1000]
#define WS_PSCORE 1829888u                    // f32[8][4096]
#define WS_PRAW   1960960u                    // f32[8][4096][4]
#define WS_POFF   2485248u                    // f32[8][4096][4]
#define WS_SSCORE 3009536u                    // f32[8][4096]
#define WS_SRAW   3140608u                    // f32[8][4096][4]
#define WS_SOFF   3664896u                    // f32[8][4096][4]
#define WS_TOTAL  4189184u

__device__ __forceinline__ int lvl_of(int a) {
  return (a < L0_END) ? 0 : (a < L1_END) ? 1 : (a < L2_END) ? 2 : 3;
}
__device__ __forceinline__ unsigned score_key(float s) {
  unsigned u = __float_as_uint(s);
  return (u & 0x80000000u) ? ~u : (u | 0x80000000u);
}

// ---------------------------------------------------------------------------
__global__ void k_init(unsigned* hist, unsigned* state, float* out) {
  int id = blockIdx.x * blockDim.x + threadIdx.x;
  int np = gridDim.x * blockDim.x;
  for (int i = id; i < 3 * 32 * 4096; i += np) hist[i] = 0u;
  for (int i = id; i < 32 * 8; i += np) state[i] = 0u;
  for (int i = id; i < N_B * K_LVL * 5; i += np) out[i] = 0.0f;
}

// ---------------------------------------------------------------------------
// histogram pass: pass0 = bits[31:20], pass1 = bits[19:8], pass2 = bits[7:0]
__global__ void k_hist(const float* __restrict__ obj, const SelState* __restrict__ st,
                       unsigned* __restrict__ histAll, int pass) {
  int id = blockIdx.x * blockDim.x + threadIdx.x;
  if (id >= N_B * A_TOTAL) return;
  if (id + 16384 < N_B * A_TOTAL)
    __builtin_prefetch(&obj[id + 16384], 0, 1);   // global_prefetch_b8
  int b = id / A_TOTAL, a = id % A_TOTAL;
  int g = b * 4 + lvl_of(a);
  unsigned key = score_key(obj[id]);
  unsigned bin;
  if (pass == 0) {
    bin = key >> 20;
  } else if (pass == 1) {
    if ((key >> 20) != st[g].prefix) return;
    bin = (key >> 8) & 0xFFFu;
  } else {
    if ((key >> 8) != st[g].prefix) return;
    bin = key & 0xFFu;
  }
  atomicAdd(&histAll[pass * 32 * 4096 + g * 4096 + bin], 1u);
}

// ---------------------------------------------------------------------------
// threshold scan: chunk sums via WMMA tensor-pipe reduction, then serial scan.
__global__ __launch_bounds__(1024) void k_pick(const unsigned* __restrict__ histAll,
                                               SelState* __restrict__ st, int pass) {
  int g = blockIdx.x;  // (b*4 + level)
  const unsigned* hist = histAll + pass * 32 * 4096 + g * 4096;
  int nbins = (pass == 2) ? 256 : 4096;
  int nch = nbins >> 6;  // 64-bin chunks
  __shared__ float csum[64];
  int lane = threadIdx.x & 31;
  // Wave id routed through readfirstlane so the chunk loop is SGPR-controlled:
  // WMMA then always executes with EXEC all-ones (ISA requirement), no lane masking.
  int wvu = __builtin_amdgcn_readfirstlane((int)(threadIdx.x >> 5));
  for (int c = wvu; c < nch; c += 32) {
    float b0 = (float)hist[c * 64 + lane];
    float b1 = (float)hist[c * 64 + 32 + lane];
    float tot;
#ifdef HAVE_WMMA_F32X4
    // D = ones(16x4) x B(4x16): every row of D = column sums of B, so
    // sum over lanes 0..15 of D-row0 = total of all 64 counts (exact in f32).
    v2f av = {1.0f, 1.0f};
    v2f bv = {b0, b1};
    v8f cv = {};
    cv = __builtin_amdgcn_wmma_f32_16x16x4_f32(false, av, false, bv,
                                               (short)0, cv, false, false);
    float v = cv[0];
    v += __shfl_xor(v, 1, 16);
    v += __shfl_xor(v, 2, 16);
    v += __shfl_xor(v, 4, 16);
    v += __shfl_xor(v, 8, 16);
    tot = v;
#else
    float v = b0 + b1;
    for (int m = 16; m; m >>= 1) v += __shfl_xor(v, m, 32);
    tot = v;
#endif
    if (lane == 0) csum[c] = tot;
  }
  __syncthreads();
  if (threadIdx.x == 0) {
    SelState s = st[g];
    unsigned kNeed = (unsigned)K_LVL - s.above;
    unsigned acc = 0, strictA = 0;
    int t = 0;
    for (int c = nch - 1; c >= 0; --c) {
      unsigned cs = (unsigned)(csum[c] + 0.5f);
      if (acc + cs >= kNeed) {
        for (int d = 63; d >= 0; --d) {
          unsigned h = hist[c * 64 + d];
          if (acc + h >= kNeed) { t = c * 64 + d; strictA = acc; break; }
          acc += h;
        }
        break;
      }
      acc += cs;
    }
    s.above += strictA;
    if (pass == 0)      s.prefix = (unsigned)t;
    else if (pass == 1) s.prefix = (s.prefix << 12) | (unsigned)t;
    else { s.T = (s.prefix << 8) | (unsigned)t; s.needEq = (unsigned)K_LVL - s.above; }
    st[g] = s;
  }
}

// ---------------------------------------------------------------------------
__global__ void k_compact(const float* __restrict__ obj, SelState* __restrict__ st,
                          float* __restrict__ candScore, unsigned* __restrict__ candIdx) {
  int id = blockIdx.x * blockDim.x + threadIdx.x;
  if (id >= N_B * A_TOTAL) return;
  int b = id / A_TOTAL, a = id % A_TOTAL;
  int g = b * 4 + lvl_of(a);
  float s = obj[id];
  unsigned key = score_key(s);
  SelState ss = st[g];
  unsigned slot;
  if (key > ss.T) {
    slot = atomicAdd(&st[g].aboveCtr, 1u);
  } else if (key == ss.T) {
    unsigned e = atomicAdd(&st[g].eqCtr, 1u);
    if (e >= ss.needEq) return;
    slot = ss.above + e;
  } else {
    return;
  }
  candScore[g * K_LVL + slot] = s;
  candIdx[g * K_LVL + slot] = (unsigned)a;
}

// ---------------------------------------------------------------------------
__global__ void k_decode(const float4* __restrict__ deltas, const float4* __restrict__ anchors,
                         const float* __restrict__ candScore, const unsigned* __restrict__ candIdx,
                         float* __restrict__ pscore, float4* __restrict__ praw,
                         float4* __restrict__ poff) {
  int id = blockIdx.x * blockDim.x + threadIdx.x;
  if (id >= N_B * N_PAD) return;
  int b = id >> 12, j = id & (N_PAD - 1);
  float sc = NEGV;
  float4 r = make_float4(0.f, 0.f, 0.f, 0.f);
  float4 o = make_float4(0.f, 0.f, 0.f, 0.f);
  if (j < K_IMG) {
    int lv = j / K_LVL, slot = j % K_LVL, g = b * 4 + lv;
    unsigned a = candIdx[g * K_LVL + slot];
    float s = candScore[g * K_LVL + slot];
    float4 anc = anchors[a];
    float4 d = deltas[b * A_TOTAL + (int)a];
    float wa = anc.z - anc.x, ha = anc.w - anc.y;
    float cxa = anc.x + 0.5f * wa, cya = anc.y + 0.5f * ha;
    float dw = fminf(d.z, BBOX_CLIP), dh = fminf(d.w, BBOX_CLIP);
    float cx = d.x * wa + cxa, cy = d.y * ha + cya;
    float w = expf(dw) * wa, h = expf(dh) * ha;
    float x1 = fminf(fmaxf(cx - 0.5f * w, 0.f), IMG_WH);
    float y1 = fminf(fmaxf(cy - 0.5f * h, 0.f), IMG_WH);
    float x2 = fminf(fmaxf(cx + 0.5f * w, 0.f), IMG_WH);
    float y2 = fminf(fmaxf(cy + 0.5f * h, 0.f), IMG_WH);
    bool valid = (x2 - x1 >= MIN_SZ) && (y2 - y1 >= MIN_SZ) && (s >= 0.0f);
    sc = valid ? s : NEGV;
    r = make_float4(x1, y1, x2, y2);
    float off = (float)lv * LVL_OFF;
    o = make_float4(x1 + off, y1 + off, x2 + off, y2 + off);
  }
  pscore[id] = sc;
  praw[id] = r;
  poff[id] = o;
}

// ---------------------------------------------------------------------------
// bitonic sort of 4096 (score,idx) descending, one block per image
__global__ __launch_bounds__(1024) void k_sort(const float* __restrict__ pscore,
                                               const float4* __restrict__ praw,
                                               const float4* __restrict__ poff,
                                               float* __restrict__ sscore,
                                               float4* __restrict__ sraw,
                                               float4* __restrict__ soff) {
  __shared__ float kk[N_PAD];
  __shared__ unsigned xx[N_PAD];
  int b = blockIdx.x;
  for (int i = threadIdx.x; i < N_PAD; i += 1024) { kk[i] = pscore[b * N_PAD + i]; xx[i] = (unsigned)i; }
  __syncthreads();
  for (int size = 2; size <= N_PAD; size <<= 1) {
    for (int stride = size >> 1; stride > 0; stride >>= 1) {
      for (int i = threadIdx.x; i < N_PAD; i += 1024) {
        int l = i ^ stride;
        if (l > i) {
          bool descSeg = (i & size) == 0;
          float ki = kk[i], kl = kk[l];
          bool sw = descSeg ? (ki < kl) : (ki > kl);
          if (sw) {
            kk[i] = kl; kk[l] = ki;
            unsigned t = xx[i]; xx[i] = xx[l]; xx[l] = t;
          }
        }
      }
      __syncthreads();
    }
  }
  for (int i = threadIdx.x; i < N_PAD; i += 1024) {
    sscore[b * N_PAD + i] = kk[i];
    unsigned s0 = xx[i];
    sraw[b * N_PAD + i] = praw[b * N_PAD + (int)s0];
    soff[b * N_PAD + i] = poff[b * N_PAD + (int)s0];
  }
}

// ---------------------------------------------------------------------------
// greedy NMS: one block per image; boxes staged to LDS via async global->LDS.
__global__ __launch_bounds__(1024) void k_nms(const float* __restrict__ sscore,
                                              const float4* __restrict__ sraw,
                                              const float4* __restrict__ soff,
                                              float* __restrict__ out_boxes,
                                              float* __restrict__ out_scores) {
  __shared__ float4 lbox[K_IMG];   // 64000 B
  __shared__ unsigned keep[125];   // 4000 bits
  __shared__ unsigned wpre[125];
  int b = blockIdx.x;
  int lane = threadIdx.x & 31, wv = threadIdx.x >> 5;

  // async stage: sorted offset-boxes -> LDS (tracked by ASYNCcnt)
  for (int j = threadIdx.x; j < K_IMG; j += 1024) {
    unsigned la = (unsigned)(uintptr_t)&lbox[j];  // low 32 bits of generic LDS ptr = LDS offset
    unsigned long long ga = (unsigned long long)(uintptr_t)&soff[b * N_PAD + j];
    asm volatile("global_load_async_to_lds_b128 %0, %1, off"
                 :: "v"(la), "v"(ga) : "memory");
  }
  asm volatile("s_wait_asynccnt 0" ::: "memory");
  __syncthreads();

  // keep init from validity (wave32 ballot -> one 32-bit word per 32 candidates)
  for (int w = wv; w < 125; w += 32) {
    int j = w * 32 + lane;
    unsigned m = (unsigned)__ballot(sscore[b * N_PAD + j] > NEGV * 0.5f);
    if (lane == 0) keep[w] = m;
  }
  __syncthreads();

  // greedy suppression over sorted order
  for (int i = 0; i < K_IMG; ++i) {
    if ((keep[i >> 5] >> (i & 31)) & 1u) {
      float4 bi = lbox[i];
      float ai = (bi.z - bi.x) * (bi.w - bi.y);
      int w0 = (i + 1) >> 5;
      for (int w = w0 + wv; w < 125; w += 32) {
        int j = w * 32 + lane;
        bool sup = false;
        if (j > i) {
          float4 bj = lbox[j];
          float aj = (bj.z - bj.x) * (bj.w - bj.y);
          float lx = fmaxf(bi.x, bj.x), ly = fmaxf(bi.y, bj.y);
          float rx = fminf(bi.z, bj.z), ry = fminf(bi.w, bj.w);
          float iw = fmaxf(rx - lx, 0.f), ih = fmaxf(ry - ly, 0.f);
          float inter = iw * ih;
          float iou = inter / (ai + aj - inter + 1e-9f);
          sup = iou > NMS_TH;
        }
        unsigned m = (unsigned)__ballot(sup);
        if (m && lane == 0) atomicAnd(&keep[w], ~m);
      }
    }
    __syncthreads();
  }

  // rank kept candidates, emit first 1000
  if (threadIdx.x == 0) {
    unsigned acc = 0;
    for (int w = 0; w < 125; ++w) { wpre[w] = acc; acc += (unsigned)__popc(keep[w]); }
  }
  __syncthreads();
  for (int j = threadIdx.x; j < K_IMG; j += 1024) {
    int w = j >> 5, bit = j & 31;
    if ((keep[w] >> bit) & 1u) {
      unsigned rank = wpre[w] + (unsigned)__popc(keep[w] & ((1u << bit) - 1u));
      if (rank < (unsigned)K_LVL) {
        float4 rb = sraw[b * N_PAD + j];
        float* ob = out_boxes + ((size_t)b * K_LVL + rank) * 4;
        ob[0] = rb.x; ob[1] = rb.y; ob[2] = rb.z; ob[3] = rb.w;
        out_scores[b * K_LVL + rank] = sscore[b * N_PAD + j];
      }
    }
  }
}

// ---------------------------------------------------------------------------
extern "C" void kernel_launch(void* const* d_in, const int* in_sizes, int n_in,
                              void* d_out, int out_size, void* d_ws, size_t ws_size,
                              hipStream_t stream) {
  (void)in_sizes; (void)n_in; (void)out_size; (void)ws_size;
  const float*  obj     = (const float*)d_in[0];
  const float4* deltas  = (const float4*)d_in[1];
  const float4* anchors = (const float4*)d_in[2];
  float* out = (float*)d_out;

  char* w = (char*)d_ws;
  unsigned* hist    = (unsigned*)(w + WS_HIST);
  SelState* st      = (SelState*)(w + WS_STATE);
  float*    cscore  = (float*)(w + WS_CSCORE);
  unsigned* cidx    = (unsigned*)(w + WS_CIDX);
  float*    pscore  = (float*)(w + WS_PSCORE);
  float4*   praw    = (float4*)(w + WS_PRAW);
  float4*   poff    = (float4*)(w + WS_POFF);
  float*    sscore  = (float*)(w + WS_SSCORE);
  float4*   sraw    = (float4*)(w + WS_SRAW);
  float4*   soff    = (float4*)(w + WS_SOFF);

  const int nItems = N_B * A_TOTAL;              // 2,088,960
  const int gItems = (nItems + 255) / 256;       // 8160

  k_init<<<512, 256, 0, stream>>>(hist, (unsigned*)st, out);
  for (int pass = 0; pass < 3; ++pass) {
    k_hist<<<gItems, 256, 0, stream>>>(obj, st, hist, pass);
    k_pick<<<32, 1024, 0, stream>>>(hist, st, pass);
  }
  k_compact<<<gItems, 256, 0, stream>>>(obj, st, cscore, cidx);
  k_decode<<<(N_B * N_PAD + 255) / 256, 256, 0, stream>>>(deltas, anchors, cscore, cidx,
                                                          pscore, praw, poff);
  k_sort<<<N_B, 1024, 0, stream>>>(pscore, praw, poff, sscore, sraw, soff);
  k_nms<<<N_B, 1024, 0, stream>>>(sscore, sraw, soff, out, out + N_B * K_LVL * 4);
}